// Lstm_81226421502219
// MI455X (gfx1250) — compile-verified
//
#include <hip/hip_runtime.h>

// 2-layer LSTM (H=6, I=2, B=4096, S=2048) + Linear(6,1) on gfx1250.
// One wave32 handles NTILE=2 independent 16-batch chains; recurrent GEMMs via
// V_WMMA_F32_16X16X4_F32 with gate columns reordered as [i|f] / [g|o] so the
// cell update needs only 2 ds_bpermute per row. Branch-free inner loop.

typedef __attribute__((ext_vector_type(2))) float v2f;
typedef __attribute__((ext_vector_type(8))) float v8f;

#define S_LEN 2048
#define B_N   4096
#define I_N   2
#define H_N   6
#define NTILE 2                      // batch tiles (chains) per wave
#define HB_STRIDE 10                 // row stride (floats) of h buffers
#define HB_SZ (16 * HB_STRIDE)       // one h buffer
#define DUMP_OFS (NTILE * 2 * HB_SZ) // per-lane dump slots after the buffers

// ---- fast transcendentals ---------------------------------------------------
__device__ __forceinline__ float tanh_exp2(float v) {
  float s = __builtin_amdgcn_rcpf(1.0f + __builtin_amdgcn_exp2f(v * -2.8853900817779268f));
  return __builtin_fmaf(2.0f, s, -1.0f);
}
#if __has_builtin(__builtin_amdgcn_tanhf)
#define TANHF(x) __builtin_amdgcn_tanhf(x)
#else
#define TANHF(x) tanh_exp2(x)
#endif

__device__ __forceinline__ float sigmoid_fast(float v) {      // uniform sigmoid
  return __builtin_fmaf(0.5f, TANHF(0.5f * v), 0.5f);
}
__device__ __forceinline__ float act_sel(float v, bool isTanh) {
  float t = TANHF(v * (isTanh ? 1.0f : 0.5f));
  return isTanh ? t : __builtin_fmaf(0.5f, t, 0.5f);
}
__device__ __forceinline__ float bperm_f(int idx_bytes, float v) {
  return __int_as_float(__builtin_amdgcn_ds_bpermute(idx_bytes, __float_as_int(v)));
}
__device__ __forceinline__ v8f wmma4(v2f a, v2f b, v8f c) {
  // D(16x16,f32) = A(16x4,f32) x B(4x16,f32) + C
  return __builtin_amdgcn_wmma_f32_16x16x4_f32(false, a, false, b, (short)0, c, false, false);
}

// Combined-B element: rows 0..5 = Whh[n][k], rows 6.. = Wih[n][k-6].
// All loads clamped in-bounds; invalid (n>=24 / k beyond) -> 0 via select.
__device__ __forceinline__ float wfetch(const float* __restrict__ Whh,
                                        const float* __restrict__ Wih,
                                        int n, int k, int in_dim) {
  int nc = (n < 24) ? n : 23;
  int kk = k - H_N;
  int kc = (kk < in_dim) ? ((kk < 0) ? 0 : kk) : (in_dim - 1);
  float vh = Whh[nc * H_N + ((k < H_N) ? k : H_N - 1)];
  float vi = Wih[nc * in_dim + kc];
  float v = (k < H_N) ? vh : ((kk < in_dim) ? vi : 0.0f);
  return (n < 24) ? v : 0.0f;
}
__device__ __forceinline__ v2f make_bpair(const float* __restrict__ Whh,
                                          const float* __restrict__ Wih,
                                          int n, int kb, int in_dim) {
  v2f r;
  r.x = wfetch(Whh, Wih, n, kb, in_dim);
  r.y = wfetch(Whh, Wih, n, kb + 1, in_dim);
  return r;
}

__global__ __launch_bounds__(32) void lstm_wmma_kernel(
    const float* __restrict__ x,
    const float* __restrict__ Wih0, const float* __restrict__ Whh0,
    const float* __restrict__ bih0, const float* __restrict__ bhh0,
    const float* __restrict__ Wih1, const float* __restrict__ Whh1,
    const float* __restrict__ bih1, const float* __restrict__ bhh1,
    const float* __restrict__ W2,   const float* __restrict__ b2,
    float* __restrict__ out)
{
  __shared__ float sm[NTILE * 2 * HB_SZ + 32];

  const int lane  = threadIdx.x;          // 0..31
  const int j     = lane & 15;
  const int hi    = lane >> 4;
  const int halfl = hi ? 16 : 0;
  const int b0    = blockIdx.x * (16 * NTILE);
  const int m     = j;                    // batch row within tile
  const int kb    = hi ? 2 : 0;           // K pair offset inside a 4-chunk
  const int rowoff = hi * (8 * HB_STRIDE);
  const int pidx  = (halfl + j + 6) << 2; // bpermute index: col j+6, same half
  const bool jlt6 = (j < 6);

  for (int t2 = lane; t2 < NTILE * 2 * HB_SZ + 32; t2 += 32) sm[t2] = 0.0f;
  float* dumpp = &sm[DUMP_OFS + lane];

  // ---- weight tiles, reordered gate columns: tile0=[i|f], tile1=[g|o] ------
  const int nt0 = j;        // gate row j      (i:0..5, f:6..11), pad j>=12
  const int nt1 = 12 + j;   // gate row 12+j   (g:12..17, o:18..23), pad j>=12
  const int nt0c = (j < 12) ? nt0 : 64;   // wfetch zero-pads n>=24
  const int nt1c = (j < 12) ? nt1 : 64;
  // layer 0: K = [h(6) | x(2)]
  v2f B0c0t0 = make_bpair(Whh0, Wih0, nt0c, 0 + kb, I_N);
  v2f B0c1t0 = make_bpair(Whh0, Wih0, nt0c, 4 + kb, I_N);
  v2f B0c0t1 = make_bpair(Whh0, Wih0, nt1c, 0 + kb, I_N);
  v2f B0c1t1 = make_bpair(Whh0, Wih0, nt1c, 4 + kb, I_N);
  // layer 1: K = [h1(6) | h0_new(6)]
  v2f B1c0t0 = make_bpair(Whh1, Wih1, nt0c, 0 + kb, H_N);
  v2f B1c1t0 = make_bpair(Whh1, Wih1, nt0c, 4 + kb, H_N);
  v2f B1c2t0 = make_bpair(Whh1, Wih1, nt0c, 8 + kb, H_N);
  v2f B1c0t1 = make_bpair(Whh1, Wih1, nt1c, 0 + kb, H_N);
  v2f B1c1t1 = make_bpair(Whh1, Wih1, nt1c, 4 + kb, H_N);
  v2f B1c2t1 = make_bpair(Whh1, Wih1, nt1c, 8 + kb, H_N);

  const int bi0 = (nt0 < 24) ? nt0 : 23;  // clamped bias indices
  const int bi1 = (nt1 < 24) ? nt1 : 23;
  const float bias_l0_t0 = (j < 12) ? (bih0[bi0] + bhh0[bi0]) : 0.0f;
  const float bias_l0_t1 = (j < 12) ? (bih0[bi1] + bhh0[bi1]) : 0.0f;
  const float bias_l1_t0 = (j < 12) ? (bih1[bi0] + bhh1[bi0]) : 0.0f;
  const float bias_l1_t1 = (j < 12) ? (bih1[bi1] + bhh1[bi1]) : 0.0f;

  float w2r[6];
  #pragma unroll
  for (int k = 0; k < 6; ++k) w2r[k] = W2[k];
  const float b2r = b2[0];

  // persistent per-chain cell state (D-layout: lane j, element r <-> c[r+8*hi][j])
  v8f c0[NTILE], c1[NTILE];
  #pragma unroll
  for (int t = 0; t < NTILE; ++t)
    #pragma unroll
    for (int r = 0; r < 8; ++r) { c0[t][r] = 0.0f; c1[t][r] = 0.0f; }

  // prefetched x
  v2f xv[NTILE];
  #pragma unroll
  for (int t = 0; t < NTILE; ++t)
    xv[t] = __builtin_nontemporal_load(
        (const v2f*)(x + ((size_t)(b0 + t * 16 + m)) * I_N));

  for (int s = 0; s < S_LEN; ++s) {
    const int sn = (s + 1 < S_LEN) ? (s + 1) : s;
    v2f xcur[NTILE];
    #pragma unroll
    for (int t = 0; t < NTILE; ++t) {
      xcur[t] = xv[t];
      xv[t] = __builtin_nontemporal_load(
          (const v2f*)(x + (((size_t)sn * B_N) + b0 + t * 16 + m) * I_N));
    }

    // ---------------- layer 0 GEMM (both chains) ----------------
    v8f g0[NTILE][2];
    #pragma unroll
    for (int t = 0; t < NTILE; ++t) {
      const float* H0 = &sm[(t * 2 + 0) * HB_SZ];
      v2f A00  = *(const v2f*)(&H0[m * HB_STRIDE + kb]);
      v2f A01l = *(const v2f*)(&H0[m * HB_STRIDE + 4]);
      v2f A01  = hi ? xcur[t] : A01l;       // upper lanes carry x[m,0..1]
      v8f a0, a1;
      #pragma unroll
      for (int r = 0; r < 8; ++r) { a0[r] = bias_l0_t0; a1[r] = bias_l0_t1; }
      a0 = wmma4(A00, B0c0t0, a0);
      a0 = wmma4(A01, B0c1t0, a0);
      a1 = wmma4(A00, B0c0t1, a1);
      a1 = wmma4(A01, B0c1t1, a1);
      g0[t][0] = a0; g0[t][1] = a1;
    }

    // ---------------- layer 0 cell update (both chains) ----------------
    #pragma unroll
    for (int t = 0; t < NTILE; ++t) {
      float* H0 = &sm[(t * 2 + 0) * HB_SZ];
      #pragma unroll
      for (int r = 0; r < 8; ++r) {
        float ia = sigmoid_fast(g0[t][0][r]);     // i at lane j, f at lane j+6
        float ga = act_sel(g0[t][1][r], jlt6);    // g at lane j, o at lane j+6
        float fg = bperm_f(pidx, ia);
        float og = bperm_f(pidx, ga);
        float cn = __builtin_fmaf(fg, c0[t][r], ia * ga);
        c0[t][r] = cn;
        float hn = og * TANHF(cn);
        float* p = jlt6 ? &H0[r * HB_STRIDE + rowoff + j] : dumpp;
        *p = hn;
      }
    }

    // ---------------- layer 1 GEMM (both chains) ----------------
    v8f g1[NTILE][2];
    #pragma unroll
    for (int t = 0; t < NTILE; ++t) {
      const float* H0 = &sm[(t * 2 + 0) * HB_SZ];
      const float* H1 = &sm[(t * 2 + 1) * HB_SZ];
      v2f A10 = *(const v2f*)(&H1[m * HB_STRIDE + kb]);
      const float* pa11 = hi ? &H0[m * HB_STRIDE + 0] : &H1[m * HB_STRIDE + 4];
      const float* pa12 = hi ? &H0[m * HB_STRIDE + 4] : &H0[m * HB_STRIDE + 2];
      v2f A11 = *(const v2f*)pa11;
      v2f A12 = *(const v2f*)pa12;
      v8f a0, a1;
      #pragma unroll
      for (int r = 0; r < 8; ++r) { a0[r] = bias_l1_t0; a1[r] = bias_l1_t1; }
      a0 = wmma4(A10, B1c0t0, a0);
      a0 = wmma4(A11, B1c1t0, a0);
      a0 = wmma4(A12, B1c2t0, a0);
      a1 = wmma4(A10, B1c0t1, a1);
      a1 = wmma4(A11, B1c1t1, a1);
      a1 = wmma4(A12, B1c2t1, a1);
      g1[t][0] = a0; g1[t][1] = a1;
    }

    // ---------------- layer 1 cell update (both chains) ----------------
    #pragma unroll
    for (int t = 0; t < NTILE; ++t) {
      float* H1 = &sm[(t * 2 + 1) * HB_SZ];
      #pragma unroll
      for (int r = 0; r < 8; ++r) {
        float ia = sigmoid_fast(g1[t][0][r]);
        float ga = act_sel(g1[t][1][r], jlt6);
        float fg = bperm_f(pidx, ia);
        float og = bperm_f(pidx, ga);
        float cn = __builtin_fmaf(fg, c1[t][r], ia * ga);
        c1[t][r] = cn;
        float hn = og * TANHF(cn);
        float* p = jlt6 ? &H1[r * HB_STRIDE + rowoff + j] : dumpp;
        *p = hn;
      }
    }

    // ---------------- output linear: out[s*B + b] = h1 . W2 + b2 ------------
    #pragma unroll
    for (int t = 0; t < NTILE; ++t) {
      const float* H1 = &sm[(t * 2 + 1) * HB_SZ];
      v2f h01 = *(const v2f*)(&H1[m * HB_STRIDE + 0]);
      v2f h23 = *(const v2f*)(&H1[m * HB_STRIDE + 2]);
      v2f h45 = *(const v2f*)(&H1[m * HB_STRIDE + 4]);
      float acc = b2r;
      acc = __builtin_fmaf(h01.x, w2r[0], acc);
      acc = __builtin_fmaf(h01.y, w2r[1], acc);
      acc = __builtin_fmaf(h23.x, w2r[2], acc);
      acc = __builtin_fmaf(h23.y, w2r[3], acc);
      acc = __builtin_fmaf(h45.x, w2r[4], acc);
      acc = __builtin_fmaf(h45.y, w2r[5], acc);
      // lanes 16..31 write the same value to the same address (benign dup)
      __builtin_nontemporal_store(acc, out + (size_t)s * B_N + b0 + t * 16 + m);
    }
  }
}

extern "C" void kernel_launch(void* const* d_in, const int* in_sizes, int n_in,
                              void* d_out, int out_size, void* d_ws, size_t ws_size,
                              hipStream_t stream) {
  (void)in_sizes; (void)n_in; (void)out_size; (void)d_ws; (void)ws_size;
  const float* x    = (const float*)d_in[0];
  const float* Wih0 = (const float*)d_in[1];
  const float* Whh0 = (const float*)d_in[2];
  const float* bih0 = (const float*)d_in[3];
  const float* bhh0 = (const float*)d_in[4];
  const float* Wih1 = (const float*)d_in[5];
  const float* Whh1 = (const float*)d_in[6];
  const float* bih1 = (const float*)d_in[7];
  const float* bhh1 = (const float*)d_in[8];
  const float* W2   = (const float*)d_in[9];
  const float* b2   = (const float*)d_in[10];
  float* out = (float*)d_out;

  dim3 grid(B_N / (16 * NTILE));
  dim3 block(32);
  hipLaunchKernelGGL(lstm_wmma_kernel, grid, block, 0, stream,
                     x, Wih0, Whh0, bih0, bhh0,
                     Wih1, Whh1, bih1, bhh1, W2, b2, out);
}